// TabPFNYPreEncoder_64828236366026
// MI455X (gfx1250) — compile-verified
//
#include <hip/hip_runtime.h>
#include <hip/hip_bf16.h>

typedef __attribute__((ext_vector_type(2))) float v2f;
typedef __attribute__((ext_vector_type(8))) float v8f;

#define SORT_N 1024
#define T1 512
#define EDIM 192      // reference constant E
#define BDIM 128      // reference constant B

// ---------------------------------------------------------------------------
// Kernel 1: per-column nanmean, impute, sort (bitonic, LDS), distinct-count
// scan, binary-search ranks. Writes (rank, indicator) float2 per element.
// One block per column b; y layout is (S, B) flat -> y[i*B + b].
// ---------------------------------------------------------------------------
__global__ __launch_bounds__(T1)
void rank_kernel(const float* __restrict__ y, const int* __restrict__ sepPtr,
                 float2* __restrict__ ra, int S, int B) {
  __shared__ float s[SORT_N];
  __shared__ int   da[SORT_N];
  __shared__ int   db[SORT_N];
  __shared__ float rsum[T1];
  __shared__ int   rcnt[T1];
  __shared__ float smean;

  const int b   = blockIdx.x;
  const int tid = threadIdx.x;
  const int sep = *sepPtr;
  const int n   = (sep < SORT_N) ? sep : SORT_N;

  // ---- nanmean over prefix (Inf included, NaN excluded, like jnp.nanmean)
  float lsum = 0.f; int lcnt = 0;
  for (int i = tid; i < n; i += T1) {
    float v = y[(unsigned)(i * B + b)];
    if (!__builtin_isnan(v)) { lsum += v; lcnt++; }
  }
  rsum[tid] = lsum; rcnt[tid] = lcnt;
  __syncthreads();
  for (int off = T1 / 2; off > 0; off >>= 1) {
    if (tid < off) { rsum[tid] += rsum[tid + off]; rcnt[tid] += rcnt[tid + off]; }
    __syncthreads();
  }
  if (tid == 0) smean = rsum[0] / (float)rcnt[0];
  __syncthreads();
  const float mean = smean;

  // ---- imputed prefix into LDS; pad to SORT_N with +Inf
  for (int i = tid; i < SORT_N; i += T1) {
    float v = __builtin_inff();
    if (i < n) {
      v = y[(unsigned)(i * B + b)];
      if (__builtin_isnan(v) || __builtin_isinf(v)) v = mean;
    }
    s[i] = v;
  }
  __syncthreads();

  // ---- bitonic sort of SORT_N floats in LDS
  for (int k = 2; k <= SORT_N; k <<= 1) {
    for (int j = k >> 1; j > 0; j >>= 1) {
      for (int i = tid; i < SORT_N; i += T1) {
        int ixj = i ^ j;
        if (ixj > i) {
          float x0 = s[i], x1 = s[ixj];
          bool up = ((i & k) == 0);
          if ((x0 > x1) == up) { s[i] = x1; s[ixj] = x0; }
        }
      }
      __syncthreads();
    }
  }

  // ---- is_new flags and inclusive scan: dist[j] = #distinct in s[0..j]
  for (int i = tid; i < SORT_N; i += T1)
    da[i] = (i == 0) ? 1 : ((s[i] != s[i - 1]) ? 1 : 0);
  __syncthreads();
  int* src = da; int* dst = db;
  for (int off = 1; off < SORT_N; off <<= 1) {
    for (int i = tid; i < SORT_N; i += T1)
      dst[i] = src[i] + ((i >= off) ? src[i - off] : 0);
    __syncthreads();
    int* t = src; src = dst; dst = t;
  }
  const int* dist = src;   // inclusive distinct-count

  // ---- queries: rank = #distinct sorted prefix values < v ; indicators
  for (int i = tid; i < S; i += T1) {
    float v = y[(unsigned)(i * B + b)];
    bool vnan = __builtin_isnan(v);
    bool vinf = __builtin_isinf(v);
    float ind = vnan ? -2.f : (vinf ? ((v > 0.f) ? 2.f : 4.f) : 0.f);
    float vi  = (vnan || vinf) ? mean : v;
    int lo = 0, hi = SORT_N;
    while (lo < hi) {                       // lower_bound in LDS
      int mid = (lo + hi) >> 1;
      if (s[mid] < vi) lo = mid + 1; else hi = mid;
    }
    float rank = (lo > 0) ? (float)dist[lo - 1] : 0.f;
    float2 p; p.x = rank; p.y = ind;
    ra[(unsigned)(i * B + b)] = p;
  }
}

// ---------------------------------------------------------------------------
// Kernel 2: out[m, e] = rank*W[e,0] + ind*W[e,1] + bias[e] via fp32 WMMA
// V_WMMA_F32_16X16X4_F32 (K padded 2 -> 4 with zeros).
// A 16x4 f32 layout: v0 = K0 (lanes 0-15) / K2 (lanes 16-31); v1 = K1 / K3.
// B 4x16 mirrors it; C/D 16x16: VGPR r, lane L -> M = r + 8*(L>=16), N = L%16.
// One wave per 16-row M tile; fully unrolled over E/16 = 12 N tiles.
// All global accesses use uniform-base + 32-bit offset (saddr form); output
// stores are nontemporal (write-once 201MB stream, don't pollute L2).
// ---------------------------------------------------------------------------
__global__ __launch_bounds__(256)
void wmma_out_kernel(const float2* __restrict__ ra, const float2* __restrict__ W2,
                     const float* __restrict__ bias, float* __restrict__ out,
                     int mTiles) {
  const int lane  = threadIdx.x & 31;
  const int wave  = threadIdx.x >> 5;
  const int mTile = blockIdx.x * (blockDim.x >> 5) + wave;
  if (mTile >= mTiles) return;              // uniform per wave -> EXEC stays full
  const int mBase = mTile << 4;
  const int half  = lane >> 4;              // 0: K=0/1 data, 1: K=2/3 zeros
  const int l     = lane & 15;

  float2 ap = ra[(unsigned)(mBase + l)];    // saddr + 32-bit voffset b64 load
  v2f a;
  a.x = half ? 0.f : ap.x;                  // A[M=l, K=0] = rank   (cndmask)
  a.y = half ? 0.f : ap.y;                  // A[M=l, K=1] = indicator

  const unsigned rowOff = (unsigned)(mBase + half * 8) * (unsigned)EDIM;

#pragma unroll
  for (int t = 0; t < EDIM / 16; ++t) {
    const int e = (t << 4) + l;
    float2 wv = W2[(unsigned)e];            // {W[e,0], W[e,1]} single b64 load
    v2f bm;
    bm.x = half ? 0.f : wv.x;               // B[K=0, N=l]
    bm.y = half ? 0.f : wv.y;               // B[K=1, N=l]
    float bv = bias[(unsigned)e];           // C[m, n] = bias[n], n = lane&15
    v8f c = {bv, bv, bv, bv, bv, bv, bv, bv};
    v8f d = __builtin_amdgcn_wmma_f32_16x16x4_f32(
        /*neg_a=*/false, a, /*neg_b=*/false, bm,
        /*c_mod=*/(short)0, c, /*reuse_a=*/false, /*reuse_b=*/false);
    const unsigned o0 = rowOff + (unsigned)e;
#pragma unroll
    for (int r = 0; r < 8; ++r)             // 64B-coalesced per half-wave,
      __builtin_nontemporal_store(d[r], out + o0 + (unsigned)(r * EDIM));
  }
}

// ---------------------------------------------------------------------------
extern "C" void kernel_launch(void* const* d_in, const int* in_sizes, int n_in,
                              void* d_out, int out_size, void* d_ws, size_t ws_size,
                              hipStream_t stream) {
  const float*  y    = (const float*)d_in[0];   // (S, B, 1) fp32
  const float2* W2   = (const float2*)d_in[1];  // (E, 2)    fp32, row pairs
  const float*  bias = (const float*)d_in[2];   // (E,)      fp32
  const int*    sep  = (const int*)d_in[3];     // scalar, read on device
  float*        out  = (float*)d_out;           // (S, B, E) fp32

  const int total = in_sizes[0];                // S * B
  const int B     = BDIM;                       // reference constant
  const int S     = total / B;                  // 2048

  float2* ra = (float2*)d_ws;                   // (S*B) x {rank, indicator} = 2 MB

  rank_kernel<<<B, T1, 0, stream>>>(y, sep, ra, S, B);

  const int mTiles = total / 16;                // 16384
  const int wavesPerBlock = 256 / 32;
  const int blocks = (mTiles + wavesPerBlock - 1) / wavesPerBlock;
  wmma_out_kernel<<<blocks, 256, 0, stream>>>(ra, W2, bias, out, mTiles);
}